// ChamferLoss_20066087207135
// MI455X (gfx1250) — compile-verified
//
#include <hip/hip_runtime.h>
#include <hip/hip_bf16.h>

typedef __attribute__((ext_vector_type(2))) float v2f;
typedef __attribute__((ext_vector_type(4))) float v4f;
typedef __attribute__((ext_vector_type(8))) float v8f;

#define NB     8       // batches
#define NPTS   8192    // points per cloud (N == M)
#define WAVES  8       // waves per block (wave32)
#define XPB    128     // x points per block = WAVES*16
#define XBLK   (NPTS / XPB)   // 64 x-blocks per batch
#define CHUNK  2048    // y points staged in LDS per pass
#define TILES  (CHUNK / 16)   // 128 y tiles per chunk
#define KPT    (CHUNK / 256)  // staged points per thread per chunk (8)
#define FMAXBITS 0x7f7fffffu  // bit pattern of FLT_MAX

// ---------------------------------------------------------------------------
// Init: global col-min workspace to +FLT_MAX bits (mins are clamped >= 0,
// so uint ordering == float ordering).
// ---------------------------------------------------------------------------
__global__ void chamfer_init(unsigned* __restrict__ miny, int n) {
    int i = blockIdx.x * blockDim.x + threadIdx.x;
    if (i < n) miny[i] = FMAXBITS;
}

// ---------------------------------------------------------------------------
// Main kernel: one v_wmma_f32_16x16x4_f32 per 16x16 distance tile.
//   A row m: (-2x0, -2x1, -2x2, 1)    B col n: (y0, y1, y2, ||y||^2)
//   C[v]   : ||x_M||^2 (loop-INVARIANT -> no per-tile C setup; SRC2 != VDST)
//   D = ||x||^2 - 2 x.y + ||y||^2 = squared-distance tile.
// B fragments are prefetched one tile ahead so the WMMA never waits on the
// same-iteration ds_load. Staging issues all global loads before packing.
// ---------------------------------------------------------------------------
__global__ __launch_bounds__(256) void chamfer_main(
    const float* __restrict__ X, const float* __restrict__ Y,
    float* __restrict__ minx, unsigned* __restrict__ miny)
{
    __shared__ __align__(16) float spack[CHUNK * 4];  // {y0,y1,y2,||y||^2}: 32 KB
    __shared__ unsigned scol[NPTS];                   // per-block col-min: 32 KB

    const int tid  = threadIdx.x;
    const int b    = blockIdx.x >> 6;        // batch
    const int xblk = blockIdx.x & (XBLK - 1);
    const int wave = tid >> 5;
    const int lane = tid & 31;
    const int n16  = lane & 15;
    const bool hi  = lane >= 16;
    const int xbase = xblk * XPB + wave * 16;

    const float* __restrict__ Xb = X + (size_t)b * NPTS * 3;
    const float* __restrict__ Yb = Y + (size_t)b * NPTS * 3;

    // init per-block col-min
    for (int i = tid; i < NPTS; i += 256) scol[i] = FMAXBITS;

    // ---- A fragment (16x4 f32, 2 VGPRs/lane) ----
    // lanes 0-15: {K0,K1} = {-2x0,-2x1};  lanes 16-31: {K2,K3} = {-2x2, 1}
    const int xi = xbase + n16;
    float x0 = Xb[xi * 3 + 0];
    float x1 = Xb[xi * 3 + 1];
    float x2 = Xb[xi * 3 + 2];
    float xn = x0 * x0 + x1 * x1 + x2 * x2;
    v2f A;
    A.x = hi ? (-2.0f * x2) : (-2.0f * x0);
    A.y = hi ? 1.0f         : (-2.0f * x1);

    // ---- loop-invariant C fragment: C[v] = ||x_M||^2, M = v (lo) / v+8 (hi) ----
    const int hsel = hi ? 8 : 0;
    v8f Cfrag;
#pragma unroll
    for (int v = 0; v < 8; ++v) Cfrag[v] = __shfl(xn, hsel + v, 32);

    float rmin[8];
#pragma unroll
    for (int v = 0; v < 8; ++v) rmin[v] = 3.402823466e+38f;

    const int boff = hi ? 2 : 0;  // lo lanes read {y0,y1}; hi lanes {y2,||y||^2}

    for (int chunk = 0; chunk < NPTS / CHUNK; ++chunk) {
        __syncthreads();
        // ---- stage + pack: batch all global loads first (8 b96 in flight),
        //      then pack {y0,y1,y2,||y||^2} with one ds_store_b128 per point.
        float a0[KPT], a1[KPT], a2[KPT];
#pragma unroll
        for (int k = 0; k < KPT; ++k) {
            const int g = chunk * CHUNK + k * 256 + tid;
            a0[k] = Yb[g * 3 + 0];
            a1[k] = Yb[g * 3 + 1];
            a2[k] = Yb[g * 3 + 2];
        }
#pragma unroll
        for (int k = 0; k < KPT; ++k) {
            v4f pk;
            pk.x = a0[k]; pk.y = a1[k]; pk.z = a2[k];
            pk.w = a0[k] * a0[k] + a1[k] * a1[k] + a2[k] * a2[k];
            *(v4f*)&spack[(k * 256 + tid) * 4] = pk;
        }
        __syncthreads();

        // ---- compute: prefetch B one tile ahead ----
        v2f Bv = *(const v2f*)(&spack[n16 * 4] + boff);
        for (int t = 0; t < TILES; ++t) {
            const int tn = (t + 1) & (TILES - 1);  // wraps for last iter (unused)
            v2f Bn = *(const v2f*)(&spack[(tn * 16 + n16) * 4] + boff);

            v8f D = __builtin_amdgcn_wmma_f32_16x16x4_f32(
                false, A, false, Bv, (short)0, Cfrag, false, false);

            float cmin = D[0];
            rmin[0] = fminf(rmin[0], D[0]);
#pragma unroll
            for (int v = 1; v < 8; ++v) {
                rmin[v] = fminf(rmin[v], D[v]);
                cmin    = fminf(cmin, D[v]);
            }
            // clamp once: keeps col-min >= 0 so uint-bit min ordering is exact.
            // lanes L and L+16 share column n -> both ds_min the same address.
            cmin = fmaxf(cmin, 0.0f);
            atomicMin(&scol[chunk * CHUNK + t * 16 + n16], __float_as_uint(cmin));

            Bv = Bn;
        }
    }

    // ---- finalize row mins: reduce across the 16 column-lanes, then clamp ----
#pragma unroll
    for (int v = 0; v < 8; ++v) {
        float r = rmin[v];
        r = fminf(r, __shfl_xor(r, 1, 32));
        r = fminf(r, __shfl_xor(r, 2, 32));
        r = fminf(r, __shfl_xor(r, 4, 32));
        r = fminf(r, __shfl_xor(r, 8, 32));
        rmin[v] = fmaxf(r, 0.0f);
    }
    // lane 0 holds rows M=0..7 (VGPR v -> M=v); lane 16 holds rows M=8..15
    if (lane == 0) {
#pragma unroll
        for (int v = 0; v < 8; ++v)
            minx[(size_t)b * NPTS + xbase + v] = rmin[v];
    } else if (lane == 16) {
#pragma unroll
        for (int v = 0; v < 8; ++v)
            minx[(size_t)b * NPTS + xbase + 8 + v] = rmin[v];
    }

    // ---- flush per-block col mins to global ----
    __syncthreads();
    for (int i = tid; i < NPTS; i += 256)
        atomicMin(&miny[(size_t)b * NPTS + i], scol[i]);
}

// ---------------------------------------------------------------------------
// Final scalar reduce. N == M, so loss = (sum(minx) + sum(miny)) / (B*N).
// Single block, fixed order -> deterministic.
// ---------------------------------------------------------------------------
__global__ __launch_bounds__(256) void chamfer_reduce(
    const float* __restrict__ minx, const unsigned* __restrict__ miny,
    float* __restrict__ out)
{
    __shared__ float sdata[256];
    float s = 0.0f;
    for (int i = threadIdx.x; i < NB * NPTS; i += 256)
        s += minx[i] + __uint_as_float(miny[i]);
    sdata[threadIdx.x] = s;
    __syncthreads();
    for (int off = 128; off > 0; off >>= 1) {
        if (threadIdx.x < off) sdata[threadIdx.x] += sdata[threadIdx.x + off];
        __syncthreads();
    }
    if (threadIdx.x == 0) out[0] = sdata[0] / (float)(NB * NPTS);
}

extern "C" void kernel_launch(void* const* d_in, const int* in_sizes, int n_in,
                              void* d_out, int out_size, void* d_ws, size_t ws_size,
                              hipStream_t stream) {
    const float* X = (const float*)d_in[0];
    const float* Y = (const float*)d_in[1];
    float* out = (float*)d_out;

    // workspace: minx (B*N floats) | miny (B*N uints) = 512 KB
    float*    minx = (float*)d_ws;
    unsigned* miny = (unsigned*)((char*)d_ws + (size_t)NB * NPTS * sizeof(float));

    const int total = NB * NPTS;
    chamfer_init<<<(total + 255) / 256, 256, 0, stream>>>(miny, total);
    chamfer_main<<<NB * XBLK, 256, 0, stream>>>(X, Y, minx, miny);
    chamfer_reduce<<<1, 256, 0, stream>>>(minx, miny, out);
}